// GATActor_66486093742483
// MI455X (gfx1250) — compile-verified
//
#include <hip/hip_runtime.h>
#include <hip/hip_bf16.h>

// ---------------------------------------------------------------------------
// 2-layer GAT forward for MI455X (gfx1250, wave32).
//   GEMMs         : V_WMMA_F32_16X16X4_F32, one wave per 16x16 C tile.
//   Edge softmax  : 3 passes (atomic max, atomic sum, weighted scatter-add),
//                   attention logits recomputed (2 gathered floats/head).
//   Aggregation   : one wave32 per edge; f32 global atomics (hit in 192MB L2).
// ---------------------------------------------------------------------------

typedef float v2f __attribute__((ext_vector_type(2)));
typedef float v8f __attribute__((ext_vector_type(8)));

#define N_FEAT   128
#define HIDDEN   32
#define HEADS    4
#define NCLS     32
#define NEG_SLOPE 0.2f

// ---------------- generic fill ----------------
__global__ void fill_f32(float* __restrict__ p, float v, long long n) {
  long long i = (long long)blockIdx.x * blockDim.x + threadIdx.x;
  if (i < n) p[i] = v;
}

// ---------------- fp32 WMMA GEMM: C[M,N] = A[M,K] @ B[K,N] (row major) -----
// Requires M%16==0, N%16==0, K%4==0. One wave (32 threads) per 16x16 tile.
// A frag (16x4 f32): lane L in [0,16): row=L, v0=K0,v1=K1 ; lanes [16,32): K2,K3.
// B frag (4x16 f32): same pattern with row->col.
// C/D (16x16 f32): VGPR i: lanes 0-15 -> M=i, N=lane ; lanes 16-31 -> M=i+8.
__global__ void wmma_gemm_f32(const float* __restrict__ A,
                              const float* __restrict__ B,
                              float* __restrict__ C,
                              int M, int N, int K) {
  const int lane = threadIdx.x & 31;
  const int rmod = lane & 15;
  const int koff = (lane >> 4) * 2;          // 0 or 2
  const int m = blockIdx.x * 16 + rmod;      // A row handled by this lane
  const int n = blockIdx.y * 16 + rmod;      // B col handled by this lane
  (void)M;
  v8f acc = {};
  for (int k = 0; k < K; k += 4) {
    v2f a, b;
    a.x = A[(long long)m * K + k + koff];
    a.y = A[(long long)m * K + k + koff + 1];
    b.x = B[(long long)(k + koff) * N + n];
    b.y = B[(long long)(k + koff + 1) * N + n];
    acc = __builtin_amdgcn_wmma_f32_16x16x4_f32(
        false, a, false, b, (short)0, acc, false, false);
  }
  const int crow = blockIdx.x * 16 + (lane >> 4) * 8;
  const int ccol = blockIdx.y * 16 + rmod;
#pragma unroll
  for (int i = 0; i < 8; ++i)
    C[(long long)(crow + i) * N + ccol] = acc[i];
}

// ---------------- attention score projections ------------------------------
// layer 1: a_src[n,h] = sum_c H1[n,h*32+c] * att_src[h,c]  (same for dst)
__global__ void attn_scores1(const float* __restrict__ H1,
                             const float* __restrict__ att_src,
                             const float* __restrict__ att_dst,
                             float* __restrict__ a_src,
                             float* __restrict__ a_dst, int Nn) {
  int t = blockIdx.x * blockDim.x + threadIdx.x;   // one per (node, head)
  if (t >= Nn * HEADS) return;
  int n = t >> 2, h = t & 3;
  const float* hp = H1 + (long long)n * N_FEAT + h * HIDDEN;
  const float* ws = att_src + h * HIDDEN;
  const float* wd = att_dst + h * HIDDEN;
  float ss = 0.f, sd = 0.f;
#pragma unroll
  for (int c = 0; c < HIDDEN; ++c) {
    float x = hp[c];
    ss = fmaf(x, ws[c], ss);
    sd = fmaf(x, wd[c], sd);
  }
  a_src[t] = ss;
  a_dst[t] = sd;
}

// layer 2 (heads=1): a[n] = sum_c H2[n,c] * att[c]
__global__ void attn_scores2(const float* __restrict__ H2,
                             const float* __restrict__ att_src,
                             const float* __restrict__ att_dst,
                             float* __restrict__ a_src,
                             float* __restrict__ a_dst, int Nn) {
  int n = blockIdx.x * blockDim.x + threadIdx.x;
  if (n >= Nn) return;
  const float* hp = H2 + (long long)n * NCLS;
  float ss = 0.f, sd = 0.f;
#pragma unroll
  for (int c = 0; c < NCLS; ++c) {
    float x = hp[c];
    ss = fmaf(x, att_src[c], ss);
    sd = fmaf(x, att_dst[c], sd);
  }
  a_src[n] = ss;
  a_dst[n] = sd;
}

// ---------------- atomic float max via int punning -------------------------
__device__ __forceinline__ void atomicMaxF32(float* addr, float v) {
  if (v >= 0.f)
    atomicMax((int*)addr, __float_as_int(v));
  else
    atomicMin((unsigned int*)addr, __float_as_uint(v));
}

__device__ __forceinline__ float lrelu(float v) {
  return v >= 0.f ? v : NEG_SLOPE * v;
}

__device__ __forceinline__ void edge_sd(const int* __restrict__ src,
                                        const int* __restrict__ dst,
                                        int e, int E, int& s, int& d) {
  if (e < E) { s = src[e]; d = dst[e]; }
  else       { s = e - E; d = s; }       // appended self loops
}

// ---------------- layer-1 edge passes (4 heads, float4-vectorized) ---------
__global__ void edge_max1(const int* __restrict__ src, const int* __restrict__ dst,
                          const float* __restrict__ a_src, const float* __restrict__ a_dst,
                          float* __restrict__ mx, int E, int Et) {
  int e = blockIdx.x * blockDim.x + threadIdx.x;
  if (e >= Et) return;
  int s, d; edge_sd(src, dst, e, E, s, d);
  float4 as = *(const float4*)(a_src + (long long)s * 4);
  float4 ad = *(const float4*)(a_dst + (long long)d * 4);
  float* m = mx + (long long)d * 4;
  atomicMaxF32(m + 0, lrelu(as.x + ad.x));
  atomicMaxF32(m + 1, lrelu(as.y + ad.y));
  atomicMaxF32(m + 2, lrelu(as.z + ad.z));
  atomicMaxF32(m + 3, lrelu(as.w + ad.w));
}

__global__ void edge_den1(const int* __restrict__ src, const int* __restrict__ dst,
                          const float* __restrict__ a_src, const float* __restrict__ a_dst,
                          const float* __restrict__ mx, float* __restrict__ den,
                          int E, int Et) {
  int e = blockIdx.x * blockDim.x + threadIdx.x;
  if (e >= Et) return;
  int s, d; edge_sd(src, dst, e, E, s, d);
  float4 as = *(const float4*)(a_src + (long long)s * 4);
  float4 ad = *(const float4*)(a_dst + (long long)d * 4);
  float4 m  = *(const float4*)(mx    + (long long)d * 4);
  float* dn = den + (long long)d * 4;
  atomicAdd(dn + 0, __expf(lrelu(as.x + ad.x) - m.x));
  atomicAdd(dn + 1, __expf(lrelu(as.y + ad.y) - m.y));
  atomicAdd(dn + 2, __expf(lrelu(as.z + ad.z) - m.z));
  atomicAdd(dn + 3, __expf(lrelu(as.w + ad.w) - m.w));
}

// one wave32 per edge; lane i moves cols [4i, 4i+3]; head = lane/8.
__global__ void edge_agg1(const int* __restrict__ src, const int* __restrict__ dst,
                          const float* __restrict__ a_src, const float* __restrict__ a_dst,
                          const float* __restrict__ mx, const float* __restrict__ den,
                          const float* __restrict__ H1, float* __restrict__ agg,
                          int E, int Et) {
  int wave = (blockIdx.x * blockDim.x + threadIdx.x) >> 5;
  int lane = threadIdx.x & 31;
  if (wave >= Et) return;
  int s, d; edge_sd(src, dst, wave, E, s, d);
  int h = lane >> 3;
  float v = lrelu(a_src[(long long)s * 4 + h] + a_dst[(long long)d * 4 + h]);
  float alpha = __expf(v - mx[(long long)d * 4 + h]) /
                (den[(long long)d * 4 + h] + 1e-16f);
  float4 hv = ((const float4*)(H1 + (long long)s * N_FEAT))[lane];
  float* out = agg + (long long)d * N_FEAT + lane * 4;
  atomicAdd(out + 0, alpha * hv.x);
  atomicAdd(out + 1, alpha * hv.y);
  atomicAdd(out + 2, alpha * hv.z);
  atomicAdd(out + 3, alpha * hv.w);
}

// ---------------- bias + relu (in place, layer-1 output -> layer-2 input) --
__global__ void bias_relu(float* __restrict__ p, const float* __restrict__ b,
                          long long n) {
  long long i = (long long)blockIdx.x * blockDim.x + threadIdx.x;
  if (i >= n) return;
  float v = p[i] + b[i & (N_FEAT - 1)];
  p[i] = v > 0.f ? v : 0.f;
}

// ---------------- layer-2 edge passes (1 head) -----------------------------
__global__ void edge_max2(const int* __restrict__ src, const int* __restrict__ dst,
                          const float* __restrict__ a_src, const float* __restrict__ a_dst,
                          float* __restrict__ mx, int E, int Et) {
  int e = blockIdx.x * blockDim.x + threadIdx.x;
  if (e >= Et) return;
  int s, d; edge_sd(src, dst, e, E, s, d);
  atomicMaxF32(mx + d, lrelu(a_src[s] + a_dst[d]));
}

__global__ void edge_den2(const int* __restrict__ src, const int* __restrict__ dst,
                          const float* __restrict__ a_src, const float* __restrict__ a_dst,
                          const float* __restrict__ mx, float* __restrict__ den,
                          int E, int Et) {
  int e = blockIdx.x * blockDim.x + threadIdx.x;
  if (e >= Et) return;
  int s, d; edge_sd(src, dst, e, E, s, d);
  atomicAdd(den + d, __expf(lrelu(a_src[s] + a_dst[d]) - mx[d]));
}

// one wave32 per edge; lane = class column.
__global__ void edge_agg2(const int* __restrict__ src, const int* __restrict__ dst,
                          const float* __restrict__ a_src, const float* __restrict__ a_dst,
                          const float* __restrict__ mx, const float* __restrict__ den,
                          const float* __restrict__ H2, float* __restrict__ agg,
                          int E, int Et) {
  int wave = (blockIdx.x * blockDim.x + threadIdx.x) >> 5;
  int lane = threadIdx.x & 31;
  if (wave >= Et) return;
  int s, d; edge_sd(src, dst, wave, E, s, d);
  float v = lrelu(a_src[s] + a_dst[d]);
  float alpha = __expf(v - mx[d]) / (den[d] + 1e-16f);
  atomicAdd(agg + (long long)d * NCLS + lane,
            alpha * H2[(long long)s * NCLS + lane]);
}

// ---------------- final bias + row softmax over 32 classes -----------------
// one wave32 per node, lane = class; shfl-xor butterflies for max & sum.
__global__ void softmax_out(const float* __restrict__ agg, const float* __restrict__ b,
                            float* __restrict__ out, int Nn) {
  int wave = (blockIdx.x * blockDim.x + threadIdx.x) >> 5;
  int lane = threadIdx.x & 31;
  if (wave >= Nn) return;
  float v = agg[(long long)wave * NCLS + lane] + b[lane];
  float m = v;
#pragma unroll
  for (int o = 16; o > 0; o >>= 1) m = fmaxf(m, __shfl_xor(m, o, 32));
  float e = __expf(v - m);
  float sum = e;
#pragma unroll
  for (int o = 16; o > 0; o >>= 1) sum += __shfl_xor(sum, o, 32);
  out[(long long)wave * NCLS + lane] = e / sum;
}

// ---------------------------------------------------------------------------
extern "C" void kernel_launch(void* const* d_in, const int* in_sizes, int n_in,
                              void* d_out, int out_size, void* d_ws, size_t ws_size,
                              hipStream_t stream) {
  const float* x        = (const float*)d_in[0];
  const int*   edge     = (const int*)d_in[1];
  const float* W1       = (const float*)d_in[2];
  const float* att_src1 = (const float*)d_in[3];
  const float* att_dst1 = (const float*)d_in[4];
  const float* b1       = (const float*)d_in[5];
  const float* W2       = (const float*)d_in[6];
  const float* att_src2 = (const float*)d_in[7];
  const float* att_dst2 = (const float*)d_in[8];
  const float* b2       = (const float*)d_in[9];
  float*       out      = (float*)d_out;

  const int Nn = in_sizes[0] / N_FEAT;       // 50000 (multiple of 16)
  const int E  = in_sizes[1] / 2;            // 1,600,000
  const int Et = E + Nn;                     // + self loops
  const int* srcp = edge;
  const int* dstp = edge + E;

  // workspace carve-up (floats)
  float* ws   = (float*)d_ws;
  float* H1   = ws; ws += (long long)Nn * N_FEAT;   // x @ W1            [N,128]
  float* as1  = ws; ws += (long long)Nn * HEADS;    // a_src layer1      [N,4]
  float* ad1  = ws; ws += (long long)Nn * HEADS;    // a_dst layer1      [N,4]
  float* mx1  = ws; ws += (long long)Nn * HEADS;    // segment max       [N,4]
  float* den1 = ws; ws += (long long)Nn * HEADS;    // segment sum       [N,4]
  float* agg1 = ws; ws += (long long)Nn * N_FEAT;   // layer1 out        [N,128]
  float* H2   = ws; ws += (long long)Nn * NCLS;     // h @ W2            [N,32]
  float* as2  = ws; ws += (long long)Nn;
  float* ad2  = ws; ws += (long long)Nn;
  float* mx2  = ws; ws += (long long)Nn;
  float* den2 = ws; ws += (long long)Nn;
  float* agg2 = ws; ws += (long long)Nn * NCLS;     // layer2 out        [N,32]

  const int TB = 256;
  auto blk1 = [&](long long n) { return (unsigned)((n + TB - 1) / TB); };

  // ----- layer 1 -----
  wmma_gemm_f32<<<dim3(Nn / 16, N_FEAT / 16), 32, 0, stream>>>(x, W1, H1, Nn, N_FEAT, N_FEAT);
  attn_scores1<<<blk1((long long)Nn * HEADS), TB, 0, stream>>>(H1, att_src1, att_dst1, as1, ad1, Nn);

  fill_f32<<<blk1((long long)Nn * HEADS), TB, 0, stream>>>(mx1, -1e30f, (long long)Nn * HEADS);
  fill_f32<<<blk1((long long)Nn * HEADS), TB, 0, stream>>>(den1, 0.f, (long long)Nn * HEADS);
  fill_f32<<<blk1((long long)Nn * N_FEAT), TB, 0, stream>>>(agg1, 0.f, (long long)Nn * N_FEAT);

  edge_max1<<<blk1(Et), TB, 0, stream>>>(srcp, dstp, as1, ad1, mx1, E, Et);
  edge_den1<<<blk1(Et), TB, 0, stream>>>(srcp, dstp, as1, ad1, mx1, den1, E, Et);
  edge_agg1<<<blk1((long long)Et * 32), TB, 0, stream>>>(srcp, dstp, as1, ad1, mx1, den1, H1, agg1, E, Et);

  bias_relu<<<blk1((long long)Nn * N_FEAT), TB, 0, stream>>>(agg1, b1, (long long)Nn * N_FEAT);

  // ----- layer 2 -----
  wmma_gemm_f32<<<dim3(Nn / 16, NCLS / 16), 32, 0, stream>>>(agg1, W2, H2, Nn, NCLS, N_FEAT);
  attn_scores2<<<blk1(Nn), TB, 0, stream>>>(H2, att_src2, att_dst2, as2, ad2, Nn);

  fill_f32<<<blk1(Nn), TB, 0, stream>>>(mx2, -1e30f, Nn);
  fill_f32<<<blk1(Nn), TB, 0, stream>>>(den2, 0.f, Nn);
  fill_f32<<<blk1((long long)Nn * NCLS), TB, 0, stream>>>(agg2, 0.f, (long long)Nn * NCLS);

  edge_max2<<<blk1(Et), TB, 0, stream>>>(srcp, dstp, as2, ad2, mx2, E, Et);
  edge_den2<<<blk1(Et), TB, 0, stream>>>(srcp, dstp, as2, ad2, mx2, den2, E, Et);
  edge_agg2<<<blk1((long long)Et * 32), TB, 0, stream>>>(srcp, dstp, as2, ad2, mx2, den2, H2, agg2, E, Et);

  softmax_out<<<blk1((long long)Nn * 32), TB, 0, stream>>>(agg2, b2, out, Nn);

  (void)n_in; (void)out_size; (void)ws_size;
}